// SoftDecisionTree_73693048864895
// MI455X (gfx1250) — compile-verified
//
#include <hip/hip_runtime.h>

// ---------------------------------------------------------------------------
// Soft Decision Tree loss on MI455X (gfx1250, wave32).
//
// B=16384 rows, D=512, 1023 inner nodes, 1024 leaves, OUT=18.
// Dominant op: fp32 GEMM 16384x512x1023 (~17.2 GFLOP) -> memory-bound on
// MI455X (23.3 TB/s HBM, ~100 MB traffic, p_all fits the 192 MB L2), so we
// keep reference f32 precision and use V_WMMA_F32_16X16X4_F32, with
// double-buffered GLOBAL_LOAD_ASYNC_TO_LDS_B128 staging (ASYNCcnt) when the
// toolchain exposes it.
// ---------------------------------------------------------------------------

typedef float v2f __attribute__((ext_vector_type(2)));
typedef float v8f __attribute__((ext_vector_type(8)));
typedef int   b128_t __attribute__((vector_size(16)));   // matches builtin param

#define GLOBAL_AS __attribute__((address_space(1)))
#define LDS_AS    __attribute__((address_space(3)))

#define B_ROWS   16384
#define DIM      512
#define N_INNER  1023
#define N_LEAF   1024
#define OUT_DIM  18
#define PSTRIDE  1024   // p_all row stride (1023 padded to 1024)

// workspace layout (in floats)
#define NLOGQ_OFF 0                   // 1024*18 = 18432 floats
#define NUM_OFF   18432               // 1024
#define DEN_OFF   (18432 + 1024)     // 1024
#define LOSS_OFF  (18432 + 2048)     // 16 (only [0] used)
#define P_OFF     32768               // 16384*1024 floats

#if __has_builtin(__builtin_amdgcn_global_load_async_to_lds_b128) && \
    __has_builtin(__builtin_amdgcn_s_wait_asynccnt)
#define HAVE_ASYNC_LDS 1
#endif

// copy 16 bytes global -> LDS (async DMA when available)
__device__ __forceinline__ void copy16_to_lds(const float* g, float* l) {
#ifdef HAVE_ASYNC_LDS
    __builtin_amdgcn_global_load_async_to_lds_b128(
        (GLOBAL_AS b128_t*)g, (LDS_AS b128_t*)l, 0, 0);
#else
    *(float4*)l = *(const float4*)g;
#endif
}

__device__ __forceinline__ void async_drain() {
#ifdef HAVE_ASYNC_LDS
    __builtin_amdgcn_s_wait_asynccnt(0);
#endif
}

// ---------------------------------------------------------------------------
// 0) zero the cross-kernel accumulators (num[1024], den[1024], loss[16])
// ---------------------------------------------------------------------------
__global__ void sdt_zero_kernel(float* acc) {
    for (int j = threadIdx.x; j < 2064; j += 256) acc[j] = 0.0f;
}

// ---------------------------------------------------------------------------
// 1) nlogQ[leaf][c] = -log_softmax(leaf_param[leaf])[c]   (1024 x 18)
// ---------------------------------------------------------------------------
__global__ void sdt_logq_kernel(const float* __restrict__ leaf_param,
                                float* __restrict__ nlogq) {
    int leaf = blockIdx.x * blockDim.x + threadIdx.x;
    if (leaf >= N_LEAF) return;
    const float* v = leaf_param + leaf * OUT_DIM;
    float m = v[0];
    #pragma unroll
    for (int c = 1; c < OUT_DIM; ++c) m = fmaxf(m, v[c]);
    float s = 0.0f;
    #pragma unroll
    for (int c = 0; c < OUT_DIM; ++c) s += __expf(v[c] - m);
    float ls = __logf(s);
    #pragma unroll
    for (int c = 0; c < OUT_DIM; ++c)
        nlogq[leaf * OUT_DIM + c] = -(v[c] - m - ls);
}

// ---------------------------------------------------------------------------
// 2) GEMM + sigmoid epilogue:
//    p_all[row][node] = sigmoid(beta[node] * (x[row] . W[node] + b[node]))
//    64x64 tile / block, 8 waves, V_WMMA_F32_16X16X4_F32.
//    K staged in 32-wide chunks, double-buffered: async-prefetch chunk c+1
//    while WMMAs consume chunk c.  LDS row stride 36 floats => 16B-aligned
//    staging and conflict-free b64 fragment reads (banks 4t+k, full 64-bank
//    coverage per wave access).
// ---------------------------------------------------------------------------
__global__ __launch_bounds__(256)
void sdt_gemm_kernel(const float* __restrict__ x,
                     const float* __restrict__ W,
                     const float* __restrict__ b,
                     const float* __restrict__ beta,
                     float* __restrict__ p_all) {
    __shared__ float a_s[2][64][36];   // 18 KB
    __shared__ float b_s[2][64][36];   // 18 KB

    const int tid    = threadIdx.x;
    const int lane   = tid & 31;
    const int wave   = tid >> 5;        // 0..7
    const int wave_m = wave & 3;        // 16-row band within 64
    const int wave_n = wave >> 2;       // 32-col band within 64

    const int mBase = blockIdx.x * 64;  // grid.x = 256
    const int nBase = blockIdx.y * 64;  // grid.y = 16

    // staging map: thread -> rows {r0, r0+32}, 4-float column group c4
    const int r0 = tid >> 3;            // 0..31
    const int c4 = (tid & 7) * 4;       // 0..28

    v8f acc0 = {};
    v8f acc1 = {};

    const int am0  = wave_m * 16 + (lane & 15);
    const int bn0  = wave_n * 32 + (lane & 15);
    const int kofs = (lane >> 4) << 1;  // 0 or 2 (K split across lane halves)

    // ---- stage chunk 0 into buffer 0 ----
    #pragma unroll
    for (int rr = 0; rr < 2; ++rr) {
        const int row  = r0 + rr * 32;
        const int wrow = (nBase + row > N_INNER - 1) ? (N_INNER - 1) : (nBase + row);
        copy16_to_lds(x + (size_t)(mBase + row) * DIM + c4, &a_s[0][row][c4]);
        copy16_to_lds(W + (size_t)wrow * DIM + c4,          &b_s[0][row][c4]);
    }
    async_drain();
    __syncthreads();

    for (int c = 0; c < DIM / 32; ++c) {            // 16 chunks
        const int buf = c & 1;
        if (c + 1 < DIM / 32) {
            const int kBase = (c + 1) * 32;
            #pragma unroll
            for (int rr = 0; rr < 2; ++rr) {
                const int row  = r0 + rr * 32;
                const int wrow = (nBase + row > N_INNER - 1) ? (N_INNER - 1)
                                                             : (nBase + row);
                copy16_to_lds(x + (size_t)(mBase + row) * DIM + kBase + c4,
                              &a_s[buf ^ 1][row][c4]);
                copy16_to_lds(W + (size_t)wrow * DIM + kBase + c4,
                              &b_s[buf ^ 1][row][c4]);
            }
        }

        #pragma unroll
        for (int ks = 0; ks < 8; ++ks) {
            const int k = ks * 4 + kofs;
            v2f av  = *(const v2f*)&a_s[buf][am0][k];
            v2f bv0 = *(const v2f*)&b_s[buf][bn0][k];
            v2f bv1 = *(const v2f*)&b_s[buf][bn0 + 16][k];
            acc0 = __builtin_amdgcn_wmma_f32_16x16x4_f32(
                false, av, false, bv0, (short)0, acc0, false, false);
            acc1 = __builtin_amdgcn_wmma_f32_16x16x4_f32(
                false, av, false, bv1, (short)0, acc1, false, false);
        }

        if (c + 1 < DIM / 32) async_drain();   // chunk c+1 landed in LDS
        __syncthreads();                       // all waves done with buf
    }

    // epilogue: C/D layout -> M = r + 8*(lane>=16), N = lane%16
    const int ncol0 = nBase + wave_n * 32 + (lane & 15);
    const int ncol1 = ncol0 + 16;
    const int rowb  = mBase + wave_m * 16 + ((lane >> 4) << 3);

    float b0 = 0.f, t0 = 0.f, b1 = 0.f, t1 = 0.f;
    if (ncol0 < N_INNER) { b0 = b[ncol0]; t0 = beta[ncol0]; }
    if (ncol1 < N_INNER) { b1 = b[ncol1]; t1 = beta[ncol1]; }

    #pragma unroll
    for (int r = 0; r < 8; ++r) {
        const size_t rbase = (size_t)(rowb + r) * PSTRIDE;
        if (ncol0 < N_INNER) {
            float z = t0 * (acc0[r] + b0);
            p_all[rbase + ncol0] = 1.0f / (1.0f + __expf(-z));
        }
        if (ncol1 < N_INNER) {
            float z = t1 * (acc1[r] + b1);
            p_all[rbase + ncol1] = 1.0f / (1.0f + __expf(-z));
        }
    }
}

// ---------------------------------------------------------------------------
// 3) tree recursion. One wave per batch row (4 waves / 128-thread block).
//    Ping-pong path buffers in LDS; per-node alpha sums reduced block-locally
//    with LDS f32 atomics (ds_add_f32), then one global f32 atomic per node
//    per block.  Leaf loss: gather from L2-resident nlogQ + shfl_xor reduce.
// ---------------------------------------------------------------------------
__global__ __launch_bounds__(128)
void sdt_tree_kernel(const float* __restrict__ p_all,
                     const int* __restrict__ target,
                     const float* __restrict__ nlogq,
                     float* __restrict__ num_g,
                     float* __restrict__ den_g,
                     float* __restrict__ loss_g) {
    __shared__ float path[2][4][N_LEAF];   // 32 KB
    __shared__ float num_s[1024];
    __shared__ float den_s[1024];

    const int tid  = threadIdx.x;
    const int wave = tid >> 5;
    const int lane = tid & 31;

    for (int j = tid; j < 1024; j += 128) { num_s[j] = 0.0f; den_s[j] = 0.0f; }
    if (lane == 0) path[0][wave][0] = 1.0f;
    __syncthreads();

    const int row = blockIdx.x * 4 + wave;
    const float* prow = p_all + (size_t)row * PSTRIDE;

    int offset = 0;
    int src = 0;
    for (int d = 1; d <= 10; ++d) {
        const int n = 1 << (d - 1);
        const int dst = src ^ 1;
        for (int jb = 0; jb < n; jb += 32) {
            const int j = jb + lane;
            if (j < n) {
                float p  = prow[offset + j];
                float pa = path[src][wave][j];
                atomicAdd(&num_s[offset + j], p * pa);   // ds_add_f32
                atomicAdd(&den_s[offset + j], pa);
                path[dst][wave][2 * j]     = pa * (1.0f - p);
                path[dst][wave][2 * j + 1] = pa * p;
            }
        }
        offset += n;
        src = dst;
        __syncthreads();
    }

    // leaf loss term: sum_leaf path[leaf] * (-logQ[leaf, target[row]])
    const int t = target[row];
    float acc = 0.0f;
    for (int leaf = lane; leaf < N_LEAF; leaf += 32)
        acc += path[src][wave][leaf] * nlogq[leaf * OUT_DIM + t];
    #pragma unroll
    for (int o = 16; o > 0; o >>= 1) acc += __shfl_xor(acc, o, 32);
    if (lane == 0) atomicAdd(loss_g, acc);

    __syncthreads();
    for (int j = tid; j < N_INNER; j += 128) {
        atomicAdd(&num_g[j], num_s[j]);
        atomicAdd(&den_g[j], den_s[j]);
    }
}

// ---------------------------------------------------------------------------
// 4) finalize: C = sum_nodes -lmbda(d)/2 * (log(alpha)+log(1-alpha));
//    out = loss/B + C.  Node j sits at depth d = 32 - clz(j+1).
// ---------------------------------------------------------------------------
__global__ __launch_bounds__(256)
void sdt_final_kernel(const float* __restrict__ num_g,
                      const float* __restrict__ den_g,
                      const float* __restrict__ loss_g,
                      float* __restrict__ out) {
    __shared__ float red[8];
    const int tid = threadIdx.x;
    float c = 0.0f;
    for (int j = tid; j < N_INNER; j += 256) {
        const int d = 32 - __builtin_clz(j + 1);       // 1..10
        const float lm = 0.1f * exp2f((float)(-d));
        const float alpha = num_g[j] / den_g[j];
        c += -lm * 0.5f * (__logf(alpha) + __logf(1.0f - alpha));
    }
    #pragma unroll
    for (int o = 16; o > 0; o >>= 1) c += __shfl_xor(c, o, 32);
    if ((tid & 31) == 0) red[tid >> 5] = c;
    __syncthreads();
    if (tid < 8) {
        float v = red[tid];
        #pragma unroll
        for (int o = 4; o > 0; o >>= 1) v += __shfl_xor(v, o, 8);
        if (tid == 0) out[0] = loss_g[0] / (float)B_ROWS + v;
    }
}

// ---------------------------------------------------------------------------
extern "C" void kernel_launch(void* const* d_in, const int* in_sizes, int n_in,
                              void* d_out, int out_size, void* d_ws, size_t ws_size,
                              hipStream_t stream) {
    const float* x      = (const float*)d_in[0];   // 16384*512
    const int*   target = (const int*)  d_in[1];   // 16384
    const float* W      = (const float*)d_in[2];   // 1023*512
    const float* b      = (const float*)d_in[3];   // 1023
    const float* beta   = (const float*)d_in[4];   // 1023
    const float* leafp  = (const float*)d_in[5];   // 1024*18

    float* ws    = (float*)d_ws;
    float* nlogq = ws + NLOGQ_OFF;
    float* num_g = ws + NUM_OFF;
    float* den_g = ws + DEN_OFF;
    float* lossg = ws + LOSS_OFF;
    float* p_all = ws + P_OFF;
    float* out   = (float*)d_out;

    sdt_zero_kernel<<<1, 256, 0, stream>>>(num_g);              // num+den+loss
    sdt_logq_kernel<<<4, 256, 0, stream>>>(leafp, nlogq);
    sdt_gemm_kernel<<<dim3(B_ROWS / 64, 16), 256, 0, stream>>>(x, W, b, beta, p_all);
    sdt_tree_kernel<<<B_ROWS / 4, 128, 0, stream>>>(p_all, target, nlogq,
                                                    num_g, den_g, lossg);
    sdt_final_kernel<<<1, 256, 0, stream>>>(num_g, den_g, lossg, out);
}